// aamsoftmax_45097156608560
// MI455X (gfx1250) — compile-verified
//
#include <hip/hip_runtime.h>
#include <math.h>

#define BB 4096
#define DD 256
#define CC 5994
#define NTILES ((CC + 15) / 16)   // 375 column tiles of 16
#define MARGIN_F 0.2f
#define SCALE_F 30.0f
#define EPS_COS 1e-7f
#define EPS_NORM 1e-12f
#define XS_STRIDE 260             // padded LDS row stride (floats) to avoid bank conflicts

typedef __attribute__((ext_vector_type(2))) float v2f;
typedef __attribute__((ext_vector_type(8))) float v8f;

// ---------------------------------------------------------------------------
// Kernel 1: inverse L2 norm of each column of W (W is [D, C] row-major)
// ---------------------------------------------------------------------------
__global__ __launch_bounds__(256) void aam_wnorm_kernel(const float* __restrict__ W,
                                                        float* __restrict__ winv) {
    int c = blockIdx.x * blockDim.x + threadIdx.x;
    if (c >= CC) return;
    float s = 0.0f;
#pragma unroll 8
    for (int k = 0; k < DD; ++k) {
        float w = W[(size_t)k * CC + c];
        s += w * w;
    }
    winv[c] = 1.0f / fmaxf(sqrtf(s), EPS_NORM);
}

// online-softmax merge: (m,s) running max / sum of exp(x-m); bi = argmax index
__device__ __forceinline__ void sm_merge(float& m, float& s, int& bi,
                                         float m2, float s2, int bi2) {
    if (m2 > m) {
        s = s * expf(m - m2) + s2;
        m = m2;
        bi = bi2;
    } else {
        s = s + s2 * expf(m2 - m);
        if (m2 == m && bi2 < bi) bi = bi2;  // first-index tie break (argmax semantics)
    }
}

// ---------------------------------------------------------------------------
// Kernel 2: fused cosine GEMM (fp32 WMMA) + margin + online softmax + argmax.
// One workgroup (8 waves, 256 threads) per 16-row stripe of x.
// ---------------------------------------------------------------------------
__global__ __launch_bounds__(256) void aam_main_kernel(const float* __restrict__ x,
                                                       const float* __restrict__ W,
                                                       const int* __restrict__ label,
                                                       const float* __restrict__ winv,
                                                       float* __restrict__ ploss,
                                                       float* __restrict__ pacc) {
    __shared__ float xs[16 * XS_STRIDE];   // raw x stripe, padded stride
    __shared__ float xinv[16];             // 1/||x_row||
    __shared__ int   lab[16];              // labels of the stripe
    __shared__ float tgt[16];              // margined+scaled target logit per row
    __shared__ float wm[8][16];            // per-wave reduced max
    __shared__ float wsv[8][16];           // per-wave reduced sum-exp
    __shared__ int   wbi[8][16];           // per-wave reduced argmax

    const int tid  = threadIdx.x;
    const int wid  = tid >> 5;
    const int lane = tid & 31;
    const int r0   = blockIdx.x * 16;

    // stage x stripe into LDS (coalesced)
#pragma unroll
    for (int i = 0; i < 16; ++i)
        xs[i * XS_STRIDE + tid] = x[(size_t)(r0 + i) * DD + tid];
    if (tid < 16) lab[tid] = label[r0 + tid];
    __syncthreads();

    // per-row inverse norms: 16 threads per row, butterfly reduce within half-wave
    {
        int r = tid >> 4, l16 = tid & 15;
        float s = 0.0f;
#pragma unroll
        for (int k = l16; k < DD; k += 16) {
            float v = xs[r * XS_STRIDE + k];
            s += v * v;
        }
        for (int off = 8; off > 0; off >>= 1) s += __shfl_xor(s, off, 16);
        if (l16 == 0) xinv[r] = 1.0f / fmaxf(sqrtf(s), EPS_NORM);
    }
    __syncthreads();

    const int half  = lane >> 4;   // 0: rows 0..7 of C-tile, 1: rows 8..15
    const int col16 = lane & 15;   // N index within tile, also A's M index

    float xin[8];
    int   lbr[8];
#pragma unroll
    for (int v = 0; v < 8; ++v) {
        int row = v + 8 * half;
        xin[v] = xinv[row];
        lbr[v] = lab[row];
    }

    // per-lane streaming softmax state for its 8 rows
    float sm_m[8], sm_s[8];
    int   sm_i[8];
#pragma unroll
    for (int v = 0; v < 8; ++v) { sm_m[v] = -3.4e38f; sm_s[v] = 0.0f; sm_i[v] = 0; }

    for (int t = wid; t < NTILES; t += 8) {
        const int col  = t * 16 + col16;
        const int colC = (col < CC) ? col : (CC - 1);   // clamp tail loads, masked below
        const float* bp = W + colC;

        v8f acc = {};
#pragma unroll 4
        for (int k0 = 0; k0 < DD; k0 += 4) {
            const int kA = k0 + 2 * half;
            // A fragment: 16x4 f32, M=col16, K={kA,kA+1}  (8B-aligned LDS read)
            v2f a = *(const v2f*)&xs[col16 * XS_STRIDE + kA];
            // B fragment: 4x16 f32, N=col16, K={kA,kA+1}
            v2f b;
            b.x = bp[(size_t)kA * CC];
            b.y = bp[(size_t)(kA + 1) * CC];
            acc = __builtin_amdgcn_wmma_f32_16x16x4_f32(
                false, a, false, b, (short)0, acc, false, false);
        }

        if (col < CC) {
            const float wic = winv[col];
#pragma unroll
            for (int v = 0; v < 8; ++v) {
                float c = acc[v] * xin[v] * wic;                       // cosine
                c = fminf(fmaxf(c, -1.0f + EPS_COS), 1.0f + EPS_COS);  // faithful clamp
                float logit = SCALE_F * c;
                if (col == lbr[v]) {
                    float lm = SCALE_F * cosf(acosf(c) + MARGIN_F);    // margin on target
                    logit = lm;
                    tgt[v + 8 * half] = lm;                            // unique writer
                }
                float m = sm_m[v];
                if (logit > m) {
                    sm_s[v] = sm_s[v] * expf(m - logit) + 1.0f;
                    sm_m[v] = logit;
                    sm_i[v] = col;
                } else {
                    sm_s[v] += expf(logit - m);
                }
            }
        }
    }

    // reduce the 16 lanes that share each row (within each half-wave)
#pragma unroll
    for (int v = 0; v < 8; ++v) {
        float m = sm_m[v], s = sm_s[v];
        int   bi = sm_i[v];
        for (int off = 1; off < 16; off <<= 1) {
            float m2 = __shfl_xor(m, off, 16);
            float s2 = __shfl_xor(s, off, 16);
            int  bi2 = __shfl_xor(bi, off, 16);
            sm_merge(m, s, bi, m2, s2, bi2);
        }
        if (col16 == 0) {
            wm [wid][v + 8 * half] = m;
            wsv[wid][v + 8 * half] = s;
            wbi[wid][v + 8 * half] = bi;
        }
    }
    __syncthreads();

    // merge the 8 waves (fixed order -> deterministic), emit stripe partials
    if (tid < 16) {
        float m = wm[0][tid], s = wsv[0][tid];
        int  bi = wbi[0][tid];
        for (int w = 1; w < 8; ++w)
            sm_merge(m, s, bi, wm[w][tid], wsv[w][tid], wbi[w][tid]);
        float loss_r = (m + logf(s)) - tgt[tid];            // -log softmax(target)
        float acc_r  = (bi == lab[tid]) ? 1.0f : 0.0f;
        for (int off = 8; off > 0; off >>= 1) {
            loss_r += __shfl_xor(loss_r, off, 16);
            acc_r  += __shfl_xor(acc_r,  off, 16);
        }
        if (tid == 0) {
            ploss[blockIdx.x] = loss_r;
            pacc [blockIdx.x] = acc_r;
        }
    }
}

// ---------------------------------------------------------------------------
// Kernel 3: deterministic tree reduction of the 256 stripe partials
// ---------------------------------------------------------------------------
__global__ __launch_bounds__(256) void aam_finalize_kernel(const float* __restrict__ ploss,
                                                           const float* __restrict__ pacc,
                                                           float* __restrict__ out) {
    __shared__ float ls[8], as_[8];
    int tid = threadIdx.x;
    float l = ploss[tid];
    float a = pacc[tid];
    for (int off = 16; off > 0; off >>= 1) {
        l += __shfl_xor(l, off, 32);
        a += __shfl_xor(a, off, 32);
    }
    if ((tid & 31) == 0) { ls[tid >> 5] = l; as_[tid >> 5] = a; }
    __syncthreads();
    if (tid == 0) {
        float L = 0.0f, A = 0.0f;
        for (int w = 0; w < 8; ++w) { L += ls[w]; A += as_[w]; }
        out[0] = L / (float)BB;
        out[1] = A / (float)BB * 100.0f;
    }
}

extern "C" void kernel_launch(void* const* d_in, const int* in_sizes, int n_in,
                              void* d_out, int out_size, void* d_ws, size_t ws_size,
                              hipStream_t stream) {
    const float* x     = (const float*)d_in[0];
    const float* W     = (const float*)d_in[1];
    const int*   label = (const int*)d_in[2];
    float*       out   = (float*)d_out;

    float* winv  = (float*)d_ws;        // CC floats
    float* ploss = winv + CC;           // 256 floats
    float* pacc  = ploss + (BB / 16);   // 256 floats

    aam_wnorm_kernel<<<(CC + 255) / 256, 256, 0, stream>>>(W, winv);
    aam_main_kernel<<<BB / 16, 256, 0, stream>>>(x, W, label, winv, ploss, pacc);
    aam_finalize_kernel<<<1, 256, 0, stream>>>(ploss, pacc, out);
}